// GL_GCNConv_9l_128h_w_44753559224350
// MI455X (gfx1250) — compile-verified
//
#include <hip/hip_runtime.h>
#include <hip/hip_bf16.h>

typedef __attribute__((ext_vector_type(2))) float v2f;
typedef __attribute__((ext_vector_type(8))) float v8f;

#define HIDDIM 128

// ---------------------------------------------------------------------------
// Normalization precompute
// ---------------------------------------------------------------------------
__global__ void k_deg_init(float* deg, int n) {
    int t = blockIdx.x * blockDim.x + threadIdx.x;
    if (t < n) deg[t] = 1.0f;  // self-loop weight
}

__global__ void k_deg_scatter(const int* __restrict__ dst,
                              const float* __restrict__ w,
                              float* deg, int E) {
    int t = blockIdx.x * blockDim.x + threadIdx.x;
    if (t < E) atomicAdd(&deg[dst[t]], w[t]);
}

__global__ void k_rsqrt_inplace(float* deg, int n) {
    int t = blockIdx.x * blockDim.x + threadIdx.x;
    if (t < n) {
        float d = deg[t];
        deg[t] = (d > 0.0f) ? rsqrtf(d) : 0.0f;
    }
}

__global__ void k_edge_norm(const int* __restrict__ src,
                            const int* __restrict__ dst,
                            const float* __restrict__ w,
                            const float* __restrict__ dinv,
                            float* __restrict__ norm, int E) {
    int t = blockIdx.x * blockDim.x + threadIdx.x;
    if (t < E) norm[t] = dinv[src[t]] * w[t] * dinv[dst[t]];
}

// ---------------------------------------------------------------------------
// In-place ELU (alpha=1), float4-vectorized. Applied once per layer between
// aggregation and the next GEMM (instead of redundantly inside the GEMM).
// ---------------------------------------------------------------------------
__global__ void k_elu_inplace(float* __restrict__ F, long long n4) {
    long long t = (long long)blockIdx.x * blockDim.x + threadIdx.x;
    if (t >= n4) return;
    float4 v = ((float4*)F)[t];
    v.x = (v.x > 0.0f) ? v.x : expm1f(v.x);
    v.y = (v.y > 0.0f) ? v.y : expm1f(v.y);
    v.z = (v.z > 0.0f) ? v.z : expm1f(v.z);
    v.w = (v.w > 0.0f) ? v.w : expm1f(v.w);
    ((float4*)F)[t] = v;
}

// ---------------------------------------------------------------------------
// Hidden-layer fp32 WMMA GEMM (specialized, cols == 128, no guards):
//   G[n_rows x 128] = A[n_rows x 128] @ W[128 x 128]
// One wave per 16x32 output strip: A fragment reused across 2 accumulators,
// 2 x V_WMMA_F32_16X16X4_F32 per K-step, 64 WMMAs per wave.
// n_rows must be a multiple of 16 (50000 = 3125*16).
// ---------------------------------------------------------------------------
__global__ void k_gemm128_wmma_f32(const float* __restrict__ A,
                                   const float* __restrict__ W,
                                   float* __restrict__ G,
                                   int n_rows) {
    int wave = (int)((blockIdx.x * blockDim.x + threadIdx.x) >> 5);
    int lane = threadIdx.x & 31;
    int rowTiles = n_rows >> 4;
    if (wave >= rowTiles * 4) return;        // 4 col-pairs of 32 cols each
    int tm = wave >> 2;
    int tp = wave & 3;

    int half = lane >> 4;                    // 0: lanes 0-15, 1: lanes 16-31
    int l16  = lane & 15;
    int koff = half * 2;                     // A/B fragment: K = 2*(lane/16)+v
    int col0 = tp * 32 + l16;                // B fragment: N = lane%16

    const float* ap = A + (size_t)(tm * 16 + l16) * HIDDIM + koff;
    const float* bp = W + (size_t)koff * HIDDIM + col0;

    v8f c0 = {}, c1 = {};
    #pragma unroll 8
    for (int k0 = 0; k0 < HIDDIM; k0 += 4) {
        v2f a  = { ap[0],  ap[1] };
        v2f b0 = { bp[0],  bp[HIDDIM] };          // cols col0,     K, K+1
        v2f b1 = { bp[16], bp[HIDDIM + 16] };     // cols col0+16
        c0 = __builtin_amdgcn_wmma_f32_16x16x4_f32(false, a, false, b0,
                                                   (short)0, c0, false, false);
        c1 = __builtin_amdgcn_wmma_f32_16x16x4_f32(false, a, false, b1,
                                                   (short)0, c1, false, false);
        ap += 4;
        bp += 4 * HIDDIM;
    }

    // C/D layout: VGPR r, lanes 0-15 -> M=r, lanes 16-31 -> M=r+8; N = lane%16
    int mbase = tm * 16 + half * 8;
    float* gp = G + (size_t)mbase * HIDDIM + col0;
    #pragma unroll
    for (int r = 0; r < 8; ++r) {
        gp[(size_t)r * HIDDIM]      = c0[r];
        gp[(size_t)r * HIDDIM + 16] = c1[r];
    }
}

// ---------------------------------------------------------------------------
// Generic small-cols WMMA GEMM (final layer, cols == 8): one 16x16 tile per
// wave with column guards on B loads / stores.
// ---------------------------------------------------------------------------
__global__ void k_gemm_small_wmma_f32(const float* __restrict__ A,
                                      const float* __restrict__ W,
                                      float* __restrict__ G,
                                      int n_rows, int cols) {
    int wave = (int)((blockIdx.x * blockDim.x + threadIdx.x) >> 5);
    int lane = threadIdx.x & 31;
    int rowTiles = n_rows >> 4;
    if (wave >= rowTiles) return;            // single col tile (cols <= 16)

    int half = lane >> 4;
    int l16  = lane & 15;
    int koff = half * 2;
    bool bvalid = (l16 < cols);

    const float* ap = A + (size_t)(wave * 16 + l16) * HIDDIM + koff;
    const float* bp = W + (size_t)koff * cols + l16;

    v8f c = {};
    #pragma unroll 8
    for (int k0 = 0; k0 < HIDDIM; k0 += 4) {
        v2f a = { ap[0], ap[1] };
        float b0 = bvalid ? bp[0]    : 0.0f;
        float b1 = bvalid ? bp[cols] : 0.0f;
        v2f b = { b0, b1 };
        c = __builtin_amdgcn_wmma_f32_16x16x4_f32(false, a, false, b,
                                                  (short)0, c, false, false);
        ap += 4;
        bp += (size_t)4 * cols;
    }

    if (bvalid) {
        int mbase = wave * 16 + half * 8;
        #pragma unroll
        for (int r = 0; r < 8; ++r)
            G[(size_t)(mbase + r) * cols + l16] = c[r];
    }
}

// ---------------------------------------------------------------------------
// Aggregation init: AGG[n,f] = bias[f] + dinv[n]^2 * G[n,f]   (self-loop term)
// ---------------------------------------------------------------------------
__global__ void k_agg_init(const float* __restrict__ G,
                           const float* __restrict__ dinv,
                           const float* __restrict__ bias,
                           float* __restrict__ AGG, int n, int cols) {
    long long t = (long long)blockIdx.x * blockDim.x + threadIdx.x;
    if (t >= (long long)n * cols) return;
    int node = (int)(t / cols);
    int f    = (int)(t - (long long)node * cols);
    float di = dinv[node];
    AGG[t] = bias[f] + di * di * G[t];
}

// ---------------------------------------------------------------------------
// Edge scatter, 128 features: one wave per edge, 4 features (float4) per lane.
// AGG[dst] += norm[e] * G[src]   (atomics resolve in L2: 25.6MB << 192MB)
// ---------------------------------------------------------------------------
__global__ void k_edge_scatter128(const int* __restrict__ src,
                                  const int* __restrict__ dst,
                                  const float* __restrict__ norm,
                                  const float* __restrict__ G,
                                  float* __restrict__ AGG, int E) {
    long long t = (long long)blockIdx.x * blockDim.x + threadIdx.x;
    int e = (int)(t >> 5);
    if (e >= E) return;
    int f = (int)(t & 31) * 4;
    int s = src[e], d = dst[e];
    float nw = norm[e];
    const float4 g = *(const float4*)(G + (size_t)s * HIDDIM + f);
    float* ap = AGG + (size_t)d * HIDDIM + f;
    atomicAdd(ap + 0, nw * g.x);
    atomicAdd(ap + 1, nw * g.y);
    atomicAdd(ap + 2, nw * g.z);
    atomicAdd(ap + 3, nw * g.w);
}

// Narrow variant for the final layer (cols == 8): thread per (edge, feature).
__global__ void k_edge_scatter_narrow(const int* __restrict__ src,
                                      const int* __restrict__ dst,
                                      const float* __restrict__ norm,
                                      const float* __restrict__ G,
                                      float* __restrict__ AGG, int E, int cols) {
    long long t = (long long)blockIdx.x * blockDim.x + threadIdx.x;
    int e = (int)(t / cols);
    if (e >= E) return;
    int f = (int)(t - (long long)e * cols);
    atomicAdd(&AGG[(size_t)dst[e] * cols + f],
              norm[e] * G[(size_t)src[e] * cols + f]);
}

// ---------------------------------------------------------------------------
// Host-side orchestration
// ---------------------------------------------------------------------------
static inline size_t align16(size_t x) { return (x + 15) & ~(size_t)15; }
static inline int cdiv(long long a, int b) { return (int)((a + b - 1) / b); }

extern "C" void kernel_launch(void* const* d_in, const int* in_sizes, int n_in,
                              void* d_out, int out_size, void* d_ws, size_t ws_size,
                              hipStream_t stream) {
    const float* x     = (const float*)d_in[0];
    const int*   eidx  = (const int*)d_in[1];   // [2][E]
    const float* ew    = (const float*)d_in[2];
    const float* W_in  = (const float*)d_in[3];
    const float* b_in  = (const float*)d_in[4];
    const float* W_hid = (const float*)d_in[5]; // [7][128][128]
    const float* b_hid = (const float*)d_in[6]; // [7][128]
    const float* W_out = (const float*)d_in[7]; // [128][8]
    const float* b_out = (const float*)d_in[8]; // [8]
    float* out = (float*)d_out;

    const int E    = in_sizes[2];
    const int N    = in_sizes[0] / HIDDIM;      // 50000 (multiple of 16)
    const int NCLS = in_sizes[8];               // 8

    const int* src = eidx;
    const int* dst = eidx + E;

    // Workspace layout
    char* w = (char*)d_ws;
    size_t off = 0;
    float* norm = (float*)(w + off); off += align16((size_t)E * 4);
    float* dinv = (float*)(w + off); off += align16((size_t)N * 4);
    float* G    = (float*)(w + off); off += (size_t)N * HIDDIM * 4;
    float* F0   = (float*)(w + off); off += (size_t)N * HIDDIM * 4;
    float* F1   = (float*)(w + off); off += (size_t)N * HIDDIM * 4;
    (void)ws_size;

    const int B = 256;

    // 1) Normalization: deg -> dinv -> per-edge norm
    k_deg_init<<<cdiv(N, B), B, 0, stream>>>(dinv, N);
    k_deg_scatter<<<cdiv(E, B), B, 0, stream>>>(dst, ew, dinv, E);
    k_rsqrt_inplace<<<cdiv(N, B), B, 0, stream>>>(dinv, N);
    k_edge_norm<<<cdiv(E, B), B, 0, stream>>>(src, dst, ew, dinv, norm, E);

    const int rowTiles = N / 16;                              // 3125
    const int gemmBlocksH = cdiv((long long)rowTiles * 4 * 32, B);
    const int gemmBlocksO = cdiv((long long)rowTiles * 32, B);
    const int initBlocksH = cdiv((long long)N * HIDDIM, B);
    const int scatBlocksH = cdiv((long long)E * 32, B);
    const long long n4 = (long long)N * HIDDIM / 4;
    const int eluBlocks = cdiv(n4, B);

    // 2) Layer 1: x @ W_in
    k_gemm128_wmma_f32<<<gemmBlocksH, B, 0, stream>>>(x, W_in, G, N);
    k_agg_init<<<initBlocksH, B, 0, stream>>>(G, dinv, b_in, F0, N, HIDDIM);
    k_edge_scatter128<<<scatBlocksH, B, 0, stream>>>(src, dst, norm, G, F0, E);

    // 3) Layers 2..8: ELU(prev) then hidden GEMM + aggregation
    float* Fin = F0;
    float* Fot = F1;
    for (int i = 0; i < 7; ++i) {
        const float* Wl = W_hid + (size_t)i * HIDDIM * HIDDIM;
        const float* bl = b_hid + (size_t)i * HIDDIM;
        k_elu_inplace<<<eluBlocks, B, 0, stream>>>(Fin, n4);
        k_gemm128_wmma_f32<<<gemmBlocksH, B, 0, stream>>>(Fin, Wl, G, N);
        k_agg_init<<<initBlocksH, B, 0, stream>>>(G, dinv, bl, Fot, N, HIDDIM);
        k_edge_scatter128<<<scatBlocksH, B, 0, stream>>>(src, dst, norm, G, Fot, E);
        float* tmp = Fin; Fin = Fot; Fot = tmp;
    }

    // 4) Output layer: ELU(prev), 128 -> NCLS, no activation on output
    k_elu_inplace<<<eluBlocks, B, 0, stream>>>(Fin, n4);
    k_gemm_small_wmma_f32<<<gemmBlocksO, B, 0, stream>>>(Fin, W_out, G, N, NCLS);
    k_agg_init<<<cdiv((long long)N * NCLS, B), B, 0, stream>>>(G, dinv, b_out, out, N, NCLS);
    k_edge_scatter_narrow<<<cdiv((long long)E * NCLS, B), B, 0, stream>>>(
        src, dst, norm, G, out, E, NCLS);
}